// FwIoU_41326175322263
// MI455X (gfx1250) — compile-verified
//
#include <hip/hip_runtime.h>

#define N_CLASS 21
#define FWIOU_EPS 1e-7f

typedef float v2f __attribute__((ext_vector_type(2)));
typedef float v8f __attribute__((ext_vector_type(8)));

// --- Kernel 0: zero the 63 global counters in d_ws (harness poisons ws) ---
__global__ void fwiou_zero_ws(unsigned* __restrict__ ws) {
    int t = threadIdx.x;
    if (t < 64) ws[t] = 0u;
}

// --- Kernel 1: streaming histogram with lane-replicated conflict-free LDS atomics ---
__global__ __launch_bounds__(256) void fwiou_hist(const int* __restrict__ pred,
                                                  const int* __restrict__ gt,
                                                  unsigned* __restrict__ gws,
                                                  int n) {
    // 3 histograms x 21 classes x 32 lane-replicas = 2016 words (8064 B)
    __shared__ unsigned hist[3 * N_CLASS * 32];
    const int tid  = threadIdx.x;
    const int lane = tid & 31;

    for (int i = tid; i < 3 * N_CLASS * 32; i += blockDim.x) hist[i] = 0u;
    __syncthreads();

    const int nchunks = n >> 2;                       // int4 chunks
    const int gid     = blockIdx.x * blockDim.x + tid;
    const int stride  = gridDim.x * blockDim.x;
    const int4* p4 = (const int4*)pred;
    const int4* g4 = (const int4*)gt;

    for (int i = gid; i < nchunks; i += stride) {
        // gfx1250 global_prefetch_b8; speculative, OOB silently dropped
        __builtin_prefetch(p4 + i + stride, 0, 1);
        __builtin_prefetch(g4 + i + stride, 0, 1);
        int4 a = p4[i];            // global_load_b128
        int4 b = g4[i];            // global_load_b128

        // lane-replicated counters: all 32 lanes hit distinct LDS banks
        int pa, ga;
        pa = a.x; ga = b.x;
        atomicAdd(&hist[(0 * N_CLASS + pa) * 32 + lane], 1u);
        atomicAdd(&hist[(1 * N_CLASS + ga) * 32 + lane], 1u);
        if (pa == ga) atomicAdd(&hist[(2 * N_CLASS + pa) * 32 + lane], 1u);

        pa = a.y; ga = b.y;
        atomicAdd(&hist[(0 * N_CLASS + pa) * 32 + lane], 1u);
        atomicAdd(&hist[(1 * N_CLASS + ga) * 32 + lane], 1u);
        if (pa == ga) atomicAdd(&hist[(2 * N_CLASS + pa) * 32 + lane], 1u);

        pa = a.z; ga = b.z;
        atomicAdd(&hist[(0 * N_CLASS + pa) * 32 + lane], 1u);
        atomicAdd(&hist[(1 * N_CLASS + ga) * 32 + lane], 1u);
        if (pa == ga) atomicAdd(&hist[(2 * N_CLASS + pa) * 32 + lane], 1u);

        pa = a.w; ga = b.w;
        atomicAdd(&hist[(0 * N_CLASS + pa) * 32 + lane], 1u);
        atomicAdd(&hist[(1 * N_CLASS + ga) * 32 + lane], 1u);
        if (pa == ga) atomicAdd(&hist[(2 * N_CLASS + pa) * 32 + lane], 1u);
    }

    // scalar tail (n % 4 != 0); n = 8.4M is divisible by 4, kept for generality
    if (gid == 0) {
        for (int i = (nchunks << 2); i < n; ++i) {
            int pa = pred[i], ga = gt[i];
            atomicAdd(&hist[(0 * N_CLASS + pa) * 32 + lane], 1u);
            atomicAdd(&hist[(1 * N_CLASS + ga) * 32 + lane], 1u);
            if (pa == ga) atomicAdd(&hist[(2 * N_CLASS + pa) * 32 + lane], 1u);
        }
    }

    __syncthreads();
    // fold 32 replicas, one global atomic per (hist,class) per workgroup
    for (int t = tid; t < 3 * N_CLASS; t += blockDim.x) {
        unsigned s = 0;
        for (int r = 0; r < 32; ++r) s += hist[t * 32 + r];
        atomicAdd(&gws[t], s);
    }
}

// --- Kernel 2: finalize; weighted-IoU dot product folded through v_wmma_f32_16x16x4_f32 ---
// A[m][k] = p[m + 16k] (p zero-padded past 21), B = all-ones (layout-invariant).
// D[m][n] = p[m] + p[m+16] for all n; per C/D layout lane l<16 holds D[0..7][l],
// lane l>=16 holds D[8..15][l-16]  =>  sum8(lane0) + sum8(lane16) = sum_c p[c].
__global__ void fwiou_finalize(const unsigned* __restrict__ ws,
                               float* __restrict__ out, int n) {
    const int lane = threadIdx.x;   // launched with exactly 32 threads: EXEC all-ones at WMMA
    float p = 0.0f;
    if (lane < N_CLASS) {
        float ph    = (float)ws[lane];
        float gh    = (float)ws[N_CLASS + lane];
        float it    = (float)ws[2 * N_CLASS + lane];
        float total = ph + gh;
        float uni   = total - it;
        float w     = total / (2.0f * (float)n);
        float iou   = (it + FWIOU_EPS) / (uni + FWIOU_EPS);
        p = w * iou;
    }
    // A 16x4 layout: lanes 0-15 hold row M=l at K=0 (VGPR0) and K=1 (VGPR1);
    // lanes 16-31 hold K=2,3 -> p[m+32], p[m+48] which are zero.
    float a1 = __shfl(p, (lane + 16) & 31, 32);   // p[l+16]
    float a0 = p;                                  // p[l]
    if (lane >= 16) { a0 = 0.0f; a1 = 0.0f; }
    v2f A; A.x = a0;  A.y = a1;
    v2f B; B.x = 1.0f; B.y = 1.0f;
    v8f C = {};
    C = __builtin_amdgcn_wmma_f32_16x16x4_f32(
            /*neg_a=*/false, A, /*neg_b=*/false, B,
            /*c_mod=*/(short)0, C, /*reuse_a=*/false, /*reuse_b=*/false);
    float s = C[0] + C[1] + C[2] + C[3] + C[4] + C[5] + C[6] + C[7];
    float s0  = __shfl(s, 0, 32);
    float s16 = __shfl(s, 16, 32);
    if (lane == 0) out[0] = s0 + s16;
}

extern "C" void kernel_launch(void* const* d_in, const int* in_sizes, int n_in,
                              void* d_out, int out_size, void* d_ws, size_t ws_size,
                              hipStream_t stream) {
    const int* pred = (const int*)d_in[0];
    const int* gt   = (const int*)d_in[1];
    unsigned*  ws   = (unsigned*)d_ws;
    float*     out  = (float*)d_out;
    const int  n    = in_sizes[0];

    fwiou_zero_ws<<<1, 64, 0, stream>>>(ws);

    int nchunks = n >> 2;
    int grid = (nchunks + 255) / 256;
    if (grid > 2048) grid = 2048;
    if (grid < 1)    grid = 1;
    fwiou_hist<<<grid, 256, 0, stream>>>(pred, gt, ws, n);

    fwiou_finalize<<<1, 32, 0, stream>>>(ws, out, n);
}